// DeepPolyReLUTansformer_20452634263558
// MI455X (gfx1250) — compile-verified
//
#include <hip/hip_runtime.h>
#include <stdint.h>

typedef float v2f __attribute__((ext_vector_type(2)));
typedef float v8f __attribute__((ext_vector_type(8)));

#define NN    8192
#define MM    8192
#define NWAVE 8
#define KW    (MM / NWAVE)      // 1024 K-columns per wave
#define SK    32                // K per pipeline stage (16 rows x 32 floats = 2KB)
#define NST   (KW / SK)         // 32 stages
#define STEPS (SK / 4)          // 8 WMMA K-steps per stage
#define SLOT  9                 // padded row stride in 16B slots (8 steps + 1 pad)
#define BUFF  (16 * SLOT * 4)   // floats per stage buffer (576 = 2304B)

__global__ __launch_bounds__(256) void deeppoly_relu_wmma(
    const float* __restrict__ bounds,   // [2, N] : l = bounds[0:N], u = bounds[N:2N]
    const float* __restrict__ weight,   // [N, M] row-major
    const float* __restrict__ bias,     // [N]
    const float* __restrict__ in_lower, // [M]
    const float* __restrict__ in_upper, // [M]
    float* __restrict__ out)            // [2, N] : lb then ub
{
    __shared__ float ldsW[NWAVE][2][BUFF];  // double-buffered W stage per wave
    __shared__ float red[NWAVE * 64];       // per-wave 16 rows x 4 partial sums

    const int tid  = threadIdx.x;
    const int lane = tid & 31;
    const int wave = tid >> 5;
    const int half = lane >> 4;         // A/C matrix half: K +{0,1} vs +{2,3}
    const int col  = lane & 15;         // A-matrix row / B,C-matrix column
    const int row0 = blockIdx.x * 16;
    const int kBegin = wave * KW;

    // ---- async producer constants: instr j covers rows 4j..4j+3, 128B/row ----
    const int pr = lane >> 3;           // sub-row 0..3 within the 4-row group
    const int ps = lane & 7;            // 16B k-chunk 0..7 (= WMMA step index)
    uint64_t gprod = (uint64_t)(uintptr_t)weight +
                     4ull * ((uint64_t)(row0 + pr) * MM + (uint64_t)(kBegin + ps * 4));
    const uint32_t lprod =
        (uint32_t)(uintptr_t)&ldsW[wave][0][0] + (uint32_t)((pr * SLOT + ps) * 16);

    const bool c0 = (col == 0), c1 = (col == 1), c2 = (col == 2), c3 = (col == 3);

    v8f acc = {0.f, 0.f, 0.f, 0.f, 0.f, 0.f, 0.f, 0.f};

    auto issue = [&](int st, int par) {
        uint64_t g = gprod + (uint64_t)st * (SK * 4);
        uint32_t l = lprod + (uint32_t)(par * (BUFF * 4));
        #pragma unroll
        for (int j = 0; j < 4; ++j) {
            asm volatile("global_load_async_to_lds_b128 %0, %1, off"
                         :: "v"(l), "v"(g) : "memory");
            g += 4ull * MM * 4ull;      // next 4 rows
            l += 4u * SLOT * 16u;       // next 4 row-slots
        }
    };

    auto consume = [&](int st, int par) {
        const float* lob = in_lower + kBegin + st * SK + 2 * half;
        const float* upb = in_upper + kBegin + st * SK + 2 * half;
        #pragma unroll
        for (int s = 0; s < STEPS; ++s) {
            // A-tile slice from LDS: row `col`, K = 4s + 2*half + {0,1}
            v2f w  = *(const v2f*)&ldsW[wave][par][(col * SLOT + s) * 4 + half * 2];
            v2f lo = *(const v2f*)(lob + s * 4);
            v2f up = *(const v2f*)(upb + s * 4);

            v2f wp, wn, b1, b2;
            wp.x = fmaxf(w.x, 0.f); wp.y = fmaxf(w.y, 0.f);
            wn.x = fminf(w.x, 0.f); wn.y = fminf(w.y, 0.f);
            // B1 (pairs with max(W,0)): col0 = in_lower[k], col2 = in_upper[k]
            b1.x = c0 ? lo.x : (c2 ? up.x : 0.f);
            b1.y = c0 ? lo.y : (c2 ? up.y : 0.f);
            // B2 (pairs with min(W,0)): col1 = in_upper[k], col3 = in_lower[k]
            b2.x = c1 ? up.x : (c3 ? lo.x : 0.f);
            b2.y = c1 ? up.y : (c3 ? lo.y : 0.f);

            acc = __builtin_amdgcn_wmma_f32_16x16x4_f32(false, wp, false, b1,
                                                        (short)0, acc, false, false);
            acc = __builtin_amdgcn_wmma_f32_16x16x4_f32(false, wn, false, b2,
                                                        (short)0, acc, false, false);
        }
    };

    // ---- software pipeline: double-buffered async HBM -> LDS -> WMMA ----
    issue(0, 0);
    int par = 0;
    for (int st = 0; st < NST; ++st) {
        if (st + 1 < NST) {
            issue(st + 1, par ^ 1);
            asm volatile("s_wait_asynccnt 0x4" ::: "memory");  // stage st landed
        } else {
            asm volatile("s_wait_asynccnt 0x0" ::: "memory");
        }
        consume(st, par);
        par ^= 1;
    }

    // Spill useful C columns (0..3) to LDS: VGPR r holds row r + 8*half, column `col`
    if (col < 4) {
        #pragma unroll
        for (int r = 0; r < 8; ++r)
            red[wave * 64 + (r + 8 * half) * 4 + col] = acc[r];
    }
    __syncthreads();

    // Deterministic cross-wave reduction of the 16x4 partial-sum block
    if (tid < 64) {
        float s = 0.f;
        #pragma unroll
        for (int w = 0; w < NWAVE; ++w) s += red[w * 64 + tid];
        red[tid] = s;   // each tid only overwrites the slot it alone reads (w==0)
    }
    __syncthreads();

    if (tid < 16) {
        const int row = row0 + tid;
        const float a = red[tid * 4 + 0];   // max(W,0) @ in_lower
        const float b = red[tid * 4 + 1];   // min(W,0) @ in_upper
        const float c = red[tid * 4 + 2];   // max(W,0) @ in_upper
        const float d = red[tid * 4 + 3];   // min(W,0) @ in_lower
        const float l  = bounds[row];
        const float u  = bounds[NN + row];
        const float bi = bias[row];

        const bool ind2 = (l >= 0.f);
        const bool ind3 = (u > 0.f) && (l < 0.f);
        const float diff  = ind3 ? (u - l) : 1.f;
        const float lmbda = ind2 ? 1.f : (ind3 ? (u / diff) : 0.f);
        const float mu    = ind3 ? (-l * u / diff) : 0.f;

        float lb = ind2 ? l : 0.f;
        float ub = ind2 ? u : (ind3 ? u : 0.f);
        const float new_l = ind2 ? (a + b + bi) : 0.f;
        const float new_u = lmbda * (c + d + bi) + mu;
        lb = fmaxf(lb, new_l);
        ub = fminf(ub, new_u);

        out[row]      = lb;
        out[NN + row] = ub;
    }
}

extern "C" void kernel_launch(void* const* d_in, const int* in_sizes, int n_in,
                              void* d_out, int out_size, void* d_ws, size_t ws_size,
                              hipStream_t stream) {
    const float* bounds   = (const float*)d_in[0];   // [2, 8192]
    const float* weight   = (const float*)d_in[1];   // [8192, 8192]
    const float* bias     = (const float*)d_in[2];   // [8192]
    const float* in_lower = (const float*)d_in[3];   // [8192]
    const float* in_upper = (const float*)d_in[4];   // [8192]
    float* out = (float*)d_out;                      // [2, 8192]

    deeppoly_relu_wmma<<<NN / 16, 256, 0, stream>>>(bounds, weight, bias,
                                                    in_lower, in_upper, out);
}